// GNNEncoder_29764123361838
// MI455X (gfx1250) — compile-verified
//
#include <hip/hip_runtime.h>

typedef __attribute__((ext_vector_type(2))) float v2f;
typedef __attribute__((ext_vector_type(8))) float v8f;

#define D 64
#define WSTRIDE 68   // padded LDS row stride (floats): banks (4n+c)%64 -> conflict-free
#define TPW_E 8      // 16-edge tiles per wave (edge kernel)
#define TPW_N 4      // 16-node tiles per wave (node kernel)

// ---------------------------------------------------------------- zero agg
__global__ void gine_zero_kernel(float4* __restrict__ p, int n4) {
    int i = blockIdx.x * blockDim.x + threadIdx.x;
    if (i < n4) p[i] = make_float4(0.f, 0.f, 0.f, 0.f);
}

// ---------------------------------------------------------------- edge tile body
template <bool FULL>
__device__ __forceinline__
void edge_tile(int tile, int E, int lo, int hi,
               const float* __restrict__ x,
               const long long* __restrict__ eidx,
               const float* __restrict__ edge_attr,
               const float* __restrict__ wlds,
               const float* __restrict__ lbias,
               float* __restrict__ agg)
{
    // per-lane src/dst for local edge `lo`
    int em = tile + lo;
    if (!FULL && em >= E) em = E - 1;
    int s_held = (int)eidx[em];
    int d_held = (int)eidx[(size_t)E + em];

    // A fragments: 16 edges x 64 K (fp32); lane lo = row M; VGPR0/1 = K = 4s+2hi, +1
    v2f a[16];
    {
        const float* ea = edge_attr + (size_t)em * D + hi * 2;
        #pragma unroll
        for (int s = 0; s < 16; ++s)
            a[s] = *(const v2f*)(ea + 4 * s);
    }

    #pragma unroll
    for (int nt = 0; nt < 4; ++nt) {
        v8f c = {};
        // B[k][n] = lin_w[nt*16+lo][k]; lane lo = col N; VGPR0/1 = K rows 2hi, 2hi+1
        const float* wb = wlds + (nt * 16 + lo) * WSTRIDE + hi * 2;
        #pragma unroll
        for (int s = 0; s < 16; ++s) {
            v2f b = *(const v2f*)(wb + 4 * s);
            c = __builtin_amdgcn_wmma_f32_16x16x4_f32(false, a[s], false, b,
                                                      (short)0, c, false, false);
        }
        const int col = nt * 16 + lo;
        const float bias = lbias[col];
        #pragma unroll
        for (int r = 0; r < 8; ++r) {
            const int eml = r + 8 * hi;          // local edge row (C: VGPR r -> M=r, r+8)
            if (FULL || (tile + eml < E)) {
                int sm = __shfl(s_held, eml, 32);
                int dm = __shfl(d_held, eml, 32);
                float v = c[r] + bias + x[(size_t)sm * D + col];
                v = fmaxf(v, 0.0f);
                atomicAdd(&agg[(size_t)dm * D + col], v);
            }
        }
    }
}

// ---------------------------------------------------------------- edge stage
// e = edge_attr @ lin_w^T + lin_b ; msg = relu(x[src] + e) ; atomicAdd agg[dst]
__global__ __launch_bounds__(128)
void gine_edge_kernel(const float* __restrict__ x,
                      const long long* __restrict__ eidx,   // [2, E] int64
                      const float* __restrict__ edge_attr,  // [E, 64]
                      const float* __restrict__ lin_w,      // [L, 64, 64]
                      const float* __restrict__ lin_b,      // [L, 64]
                      float* __restrict__ agg,              // [N, 64]
                      int layer, int E)
{
    __shared__ float wlds[D * WSTRIDE];
    const int tid = threadIdx.x;

    // stage lin_w[layer] (row-major [out][in]) into padded LDS (once per block)
    const float* wsrc = lin_w + (size_t)layer * D * D;
    for (int i = tid; i < D * D; i += 128)
        wlds[(i >> 6) * WSTRIDE + (i & 63)] = wsrc[i];
    __syncthreads();

    const int wave = tid >> 5;
    const int lane = tid & 31;
    const int lo = lane & 15;
    const int hi = lane >> 4;
    const float* lbias = lin_b + layer * D;

    const int tile0 = (blockIdx.x * 4 + wave) * (TPW_E * 16);
    if (tile0 >= E) return;

    #pragma unroll 1
    for (int t = 0; t < TPW_E; ++t) {
        const int tile = tile0 + t * 16;
        if (tile >= E) break;

        // prefetch next tile's streaming data (edge_attr row + edge indices)
        const int pf = tile + 16;
        if (pf + lo < E) {
            const float* pea = edge_attr + (size_t)(pf + lo) * D;
            __builtin_prefetch(pea, 0, 0);
            __builtin_prefetch(pea + 32, 0, 0);
            __builtin_prefetch(&eidx[pf + lo], 0, 0);
            __builtin_prefetch(&eidx[(size_t)E + pf + lo], 0, 0);
        }

        if (tile + 16 <= E)
            edge_tile<true >(tile, E, lo, hi, x, eidx, edge_attr, wlds, lbias, agg);
        else
            edge_tile<false>(tile, E, lo, hi, x, eidx, edge_attr, wlds, lbias, agg);
    }
}

// ---------------------------------------------------------------- node stage
// h = (1+eps)x + agg ; h = relu(h@w1^T + b1)@w2^T + b2 ; LayerNorm ; relu
__global__ __launch_bounds__(128)
void gine_node_kernel(const float* __restrict__ x,
                      const float* __restrict__ agg,
                      const float* __restrict__ w1, const float* __restrict__ b1,
                      const float* __restrict__ w2, const float* __restrict__ b2,
                      const float* __restrict__ eps,
                      const float* __restrict__ ln_g, const float* __restrict__ ln_b,
                      float* __restrict__ xout,
                      int layer, int N)
{
    __shared__ float w1s[D * WSTRIDE];
    __shared__ float w2s[D * WSTRIDE];
    __shared__ float hbuf[4][16 * WSTRIDE];
    __shared__ float mus[4][16];
    __shared__ float rss[4][16];

    const int tid = threadIdx.x;
    const float* w1p = w1 + (size_t)layer * D * D;
    const float* w2p = w2 + (size_t)layer * D * D;
    for (int i = tid; i < D * D; i += 128) {
        w1s[(i >> 6) * WSTRIDE + (i & 63)] = w1p[i];
        w2s[(i >> 6) * WSTRIDE + (i & 63)] = w2p[i];
    }
    __syncthreads();

    const int wave = tid >> 5;
    const int lane = tid & 31;
    const int lo = lane & 15;
    const int hi = lane >> 4;

    const int tile0 = (blockIdx.x * 4 + wave) * (TPW_N * 16);
    if (tile0 >= N) return;

    const float ep = 1.0f + eps[layer];
    float* hb = hbuf[wave];

    // per-lane LN affine params: store loop touches only f = lane and lane+32
    const float g0 = ln_g[layer * D + lane];
    const float g1 = ln_g[layer * D + lane + 32];
    const float bb0 = ln_b[layer * D + lane];
    const float bb1 = ln_b[layer * D + lane + 32];

    #pragma unroll 1
    for (int t = 0; t < TPW_N; ++t) {
        const int tile = tile0 + t * 16;
        if (tile >= N) break;

        // A fragments of h_in = (1+eps)*x + agg
        v2f a[16];
        {
            int row = tile + lo; if (row >= N) row = N - 1;
            const float* xp = x + (size_t)row * D + hi * 2;
            const float* gp = agg + (size_t)row * D + hi * 2;
            #pragma unroll
            for (int s = 0; s < 16; ++s) {
                v2f xv = *(const v2f*)(xp + 4 * s);
                v2f gv = *(const v2f*)(gp + 4 * s);
                a[s].x = ep * xv.x + gv.x;
                a[s].y = ep * xv.y + gv.y;
            }
        }

        // GEMM1 + bias + relu -> per-wave LDS tile (C->A transpose buffer)
        #pragma unroll
        for (int nt = 0; nt < 4; ++nt) {
            v8f c = {};
            const float* wb = w1s + (nt * 16 + lo) * WSTRIDE + hi * 2;
            #pragma unroll
            for (int s = 0; s < 16; ++s) {
                v2f b = *(const v2f*)(wb + 4 * s);
                c = __builtin_amdgcn_wmma_f32_16x16x4_f32(false, a[s], false, b,
                                                          (short)0, c, false, false);
            }
            const int col = nt * 16 + lo;
            const float bias = b1[layer * D + col];
            #pragma unroll
            for (int r = 0; r < 8; ++r)
                hb[(r + 8 * hi) * WSTRIDE + col] = fmaxf(c[r] + bias, 0.0f);
        }

        // re-read as A fragments (same-wave DS ops are in-order)
        v2f a2[16];
        {
            const float* hp = hb + lo * WSTRIDE + hi * 2;
            #pragma unroll
            for (int s = 0; s < 16; ++s)
                a2[s] = *(const v2f*)(hp + 4 * s);
        }

        // GEMM2 (keep all 4 N-tiles before overwriting hbuf)
        v8f c2[4];
        #pragma unroll
        for (int nt = 0; nt < 4; ++nt) {
            v8f c = {};
            const float* wb = w2s + (nt * 16 + lo) * WSTRIDE + hi * 2;
            #pragma unroll
            for (int s = 0; s < 16; ++s) {
                v2f b = *(const v2f*)(wb + 4 * s);
                c = __builtin_amdgcn_wmma_f32_16x16x4_f32(false, a2[s], false, b,
                                                          (short)0, c, false, false);
            }
            c2[nt] = c;
        }
        #pragma unroll
        for (int nt = 0; nt < 4; ++nt) {
            const int col = nt * 16 + lo;
            const float bias = b2[layer * D + col];
            #pragma unroll
            for (int r = 0; r < 8; ++r)
                hb[(r + 8 * hi) * WSTRIDE + col] = c2[nt][r] + bias;
        }

        // LayerNorm stats: lanes 0..15 each handle one node (banks 4*lane+f: conflict-free)
        if (lane < 16) {
            float sum = 0.0f, sq = 0.0f;
            #pragma unroll
            for (int f = 0; f < D; ++f) {
                float v = hb[lane * WSTRIDE + f];
                sum += v; sq += v * v;
            }
            float mu = sum * (1.0f / D);
            float var = sq * (1.0f / D) - mu * mu;
            mus[wave][lane] = mu;
            rss[wave][lane] = rsqrtf(var + 1e-5f);
        }

        // normalize + affine + relu, coalesced 128B stores
        #pragma unroll
        for (int i = 0; i < 32; ++i) {
            int id = i * 32 + lane;
            int nrow = id >> 6;      // == i>>1, uniform per iteration
            int f = id & 63;         // == lane (i even) or lane+32 (i odd)
            int grow = tile + nrow;
            if (grow < N) {
                float v = hb[nrow * WSTRIDE + f];
                v = (v - mus[wave][nrow]) * rss[wave][nrow];
                v = v * ((i & 1) ? g1 : g0) + ((i & 1) ? bb1 : bb0);
                xout[(size_t)grow * D + f] = fmaxf(v, 0.0f);
            }
        }
    }
}

// ---------------------------------------------------------------- launch
extern "C" void kernel_launch(void* const* d_in, const int* in_sizes, int n_in,
                              void* d_out, int out_size, void* d_ws, size_t ws_size,
                              hipStream_t stream) {
    const float*     x0        = (const float*)d_in[0];
    const long long* eidx      = (const long long*)d_in[1];
    const float*     edge_attr = (const float*)d_in[2];
    const float*     lin_w     = (const float*)d_in[3];
    const float*     lin_b     = (const float*)d_in[4];
    const float*     w1        = (const float*)d_in[5];
    const float*     b1        = (const float*)d_in[6];
    const float*     w2        = (const float*)d_in[7];
    const float*     b2        = (const float*)d_in[8];
    const float*     eps       = (const float*)d_in[9];
    const float*     ln_g      = (const float*)d_in[10];
    const float*     ln_b      = (const float*)d_in[11];

    const int N = in_sizes[0] / D;
    const int E = in_sizes[1] / 2;

    float* xbuf = (float*)d_ws;                      // [N, D]
    float* agg  = (float*)d_ws + (size_t)N * D;      // [N, D]
    float* xout = (float*)d_out;

    const int n4      = (N * D) / 4;
    const int zblocks = (n4 + 255) / 256;
    const int eblocks = (E + (64 * TPW_E) - 1) / (64 * TPW_E);
    const int nblocks = ((N + 15) / 16 + 4 * TPW_N - 1) / (4 * TPW_N);

    for (int l = 0; l < 3; ++l) {
        const float* xin = (l == 0) ? x0 : xbuf;
        float* xo = (l == 2) ? xout : xbuf;          // layer1 updates xbuf in place (row-local)
        gine_zero_kernel<<<zblocks, 256, 0, stream>>>((float4*)agg, n4);
        gine_edge_kernel<<<eblocks, 128, 0, stream>>>(xin, eidx, edge_attr,
                                                      lin_w, lin_b, agg, l, E);
        gine_node_kernel<<<nblocks, 128, 0, stream>>>(xin, agg, w1, b1, w2, b2,
                                                      eps, ln_g, ln_b, xo, l, N);
    }
}